// SagittariusLayers_29807073034466
// MI455X (gfx1250) — compile-verified
//
#include <hip/hip_runtime.h>
#include <hip/hip_bf16.h>

// mTAN multi-head attention with per-value-dim masked softmax, CDNA5 (gfx1250).
//
// Roofline: p_attn output is 8*8*64*256*64 f32 = 256 MB -> ~11us at 23.3 TB/s HBM.
// Total FLOPs ~5 GFLOP is negligible, so the kernel is organized around a single
// coalesced streaming write of p_attn; all GEMMs use V_WMMA_F32_16X16X4_F32.

typedef __attribute__((ext_vector_type(2))) float v2f;
typedef __attribute__((ext_vector_type(8))) float v8f;

#define NB   8      // batch N
#define TKK  256    // Tk
#define TQQ  64     // Tq
#define NH   8      // heads
#define KQD  128    // kq dim
#define VDIM 64     // value dim

__device__ __forceinline__ v8f wmma4(v2f a, v2f b, v8f c) {
  // D = A(16x4,f32) * B(4x16,f32) + C(16x16,f32)
  return __builtin_amdgcn_wmma_f32_16x16x4_f32(false, a, false, b, (short)0, c,
                                               false, false);
}

// ---------------------------------------------------------------------------
// Generic row-major f32 GEMM + bias: C[M,N] = A[M,K] @ B[K,N] + bias[N].
// One 16x16 tile per wave; 8 waves per block. M,N,K multiples of 16/16/4.
// ---------------------------------------------------------------------------
__global__ void proj_kernel(const float* __restrict__ A, const float* __restrict__ B,
                            const float* __restrict__ bias, float* __restrict__ C,
                            int M, int N, int K) {
  const int lane = threadIdx.x & 31;
  const int wave = threadIdx.x >> 5;
  const int half = lane >> 4, lh = lane & 15;
  const int tid  = blockIdx.x * 8 + wave;
  const int tilesN = N >> 4;
  const int tM = tid / tilesN, tN = tid % tilesN;
  if (tM * 16 >= M) return;            // uniform per wave
  const int m0 = tM * 16, n0 = tN * 16;

  v8f acc = {0.f, 0.f, 0.f, 0.f, 0.f, 0.f, 0.f, 0.f};
  const float* arow = A + (size_t)(m0 + lh) * K;  // A: lane holds row M=lh
  for (int k = 0; k < K; k += 4) {
    const int kk = k + 2 * half;       // lanes 0-15: K k,k+1 ; lanes 16-31: k+2,k+3
    v2f a, b;
    a.x = arow[kk];
    a.y = arow[kk + 1];
    b.x = B[(size_t)kk * N + n0 + lh]; // B: lane holds column N=lh
    b.y = B[(size_t)(kk + 1) * N + n0 + lh];
    acc = wmma4(a, b, acc);
  }
  const float bv = bias[n0 + lh];
  for (int v = 0; v < 8; ++v) {
    const int m = m0 + v + 8 * half;   // C/D: VGPR v -> M = v (+8 for upper half)
    C[(size_t)m * N + n0 + lh] = acc[v] + bv;
  }
}

// ---------------------------------------------------------------------------
// Attention core: one block per (q-chunk of 16, head h, batch n).
//  Phase 0: pack mask[n,:,:] into a 2KB LDS bitfield.
//  Phase 1: s[16,256] = hq_chunk @ hk[n,h]^T * rsqrt(128) via WMMA -> LDS.
//  Phase 2: per (q,d) masked softmax over k; stream p_attn, accumulate out.
// ---------------------------------------------------------------------------
__global__ void attn_kernel(const float* __restrict__ hq, const float* __restrict__ hk,
                            const int* __restrict__ mask, const float* __restrict__ value,
                            float* __restrict__ out_ws, float* __restrict__ p_out) {
  __shared__ float    s_lds[16 * TKK];   // 16 KB score tile
  __shared__ unsigned mbits[TKK * 2];    // 2 KB  mask bits: bit(k,d)

  const int t  = threadIdx.x;            // 0..255
  const int qc = blockIdx.x;             // 0..3   (q chunk)
  const int h  = blockIdx.y;             // 0..7
  const int n  = blockIdx.z;             // 0..7
  const int q0 = qc * 16;

  // --- Phase 0: thread t packs mask row k=t (64 ints -> 2 words) -------------
  {
    const int k = t;
    const int* mrow = mask + ((size_t)n * TKK + k) * VDIM;
    for (int j = 0; j < 2; ++j) {
      unsigned bits = 0u;
      for (int b = 0; b < 32; ++b)
        bits |= (mrow[j * 32 + b] != 0 ? 1u : 0u) << b;
      mbits[k * 2 + j] = bits;
    }
  }

  // --- Phase 1: score tile via WMMA (A = hq rows, B = hk^T columns) ----------
  {
    const int lane = t & 31, wave = t >> 5;
    const int half = lane >> 4, lh = lane & 15;
    const float* aq = hq + (size_t)(q0 + lh) * (NH * KQD) + h * KQD;
    const float scale = 0.088388347648318447f;  // 1/sqrt(128)
    for (int rep = 0; rep < 2; ++rep) {
      const int n0 = (wave + rep * 8) * 16;     // key-tile base
      const float* bkr = hk + ((size_t)n * TKK + n0 + lh) * (NH * KQD) + h * KQD;
      v8f acc = {0.f, 0.f, 0.f, 0.f, 0.f, 0.f, 0.f, 0.f};
      for (int k = 0; k < KQD; k += 4) {
        const int kk = k + 2 * half;
        v2f a, b;
        a.x = aq[kk];  a.y = aq[kk + 1];
        b.x = bkr[kk]; b.y = bkr[kk + 1];       // B^T[nIdx][c] = hk[key][c]
        acc = wmma4(a, b, acc);
      }
      for (int v = 0; v < 8; ++v)
        s_lds[(v + 8 * half) * TKK + n0 + lh] = acc[v] * scale;
    }
  }
  __syncthreads();

  // --- Phase 2: masked softmax over k per (q,d); coalesced p_attn stream -----
  const int d  = t & 63;                 // lanes contiguous in d -> coalesced
  const int qg = t >> 6;
  const unsigned wsel = (unsigned)(d >> 5);
  const unsigned bit  = 1u << (d & 31);
  for (int i = 0; i < 4; ++i) {
    const int q = qg * 4 + i;
    const float* srow = &s_lds[q * TKK];            // broadcast reads
    float mx = -3.402823466e38f;
    for (int k = 0; k < TKK; ++k) {
      const float x = srow[k] + ((mbits[k * 2 + wsel] & bit) ? 0.f : -1e9f);
      mx = fmaxf(mx, x);
    }
    float l = 0.f;
    for (int k = 0; k < TKK; ++k) {
      const float x = srow[k] + ((mbits[k * 2 + wsel] & bit) ? 0.f : -1e9f);
      l += __expf(x - mx);
    }
    const float inv = 1.0f / l;
    float o = 0.f;
    float* prow = p_out +
        ((((size_t)(n * NH + h) * TQQ + (q0 + q)) * TKK) * VDIM + d);
    const float* vrow = value + ((size_t)n * TKK) * VDIM + d;
    for (int k = 0; k < TKK; ++k) {
      const float x = srow[k] + ((mbits[k * 2 + wsel] & bit) ? 0.f : -1e9f);
      const float p = __expf(x - mx) * inv;
      prow[(size_t)k * VDIM] = p;                   // the 256 MB stream
      o += p * vrow[(size_t)k * VDIM];
    }
    out_ws[(((size_t)(n * NH + h) * TQQ) + (q0 + q)) * VDIM + d] = o;
  }
}

// ---------------------------------------------------------------------------
// Output projection: rep[n,q, h*64+d] = out_ws[n,h,q,d]; out = rep @ Wout + bout.
// M = 512 (n*64+q), N = 64, K = 512 (h*64+d). WMMA with gather on A.
// ---------------------------------------------------------------------------
__global__ void outproj_kernel(const float* __restrict__ out_ws,
                               const float* __restrict__ Wout,
                               const float* __restrict__ bout,
                               float* __restrict__ out_rep) {
  const int lane = threadIdx.x & 31, wave = threadIdx.x >> 5;
  const int half = lane >> 4, lh = lane & 15;
  const int tid = blockIdx.x * 8 + wave;   // 0..127 tiles (32 M-tiles x 4 N-tiles)
  const int tM = tid >> 2, tN = tid & 3;
  const int m0 = tM * 16, n0 = tN * 16;
  const int m  = m0 + lh;
  const int nb = m >> 6, q = m & 63;
  const float* abase = out_ws + (size_t)nb * (NH * TQQ * VDIM) + (size_t)q * VDIM;

  v8f acc = {0.f, 0.f, 0.f, 0.f, 0.f, 0.f, 0.f, 0.f};
  for (int k = 0; k < NH * VDIM; k += 4) {
    const int kk = k + 2 * half;
    const int hh = kk >> 6, dd = kk & 63;  // dd even -> dd+1 in same head block
    v2f a, b;
    const float* ap = abase + (size_t)hh * (TQQ * VDIM) + dd;
    a.x = ap[0];
    a.y = ap[1];
    b.x = Wout[(size_t)kk * VDIM + n0 + lh];
    b.y = Wout[(size_t)(kk + 1) * VDIM + n0 + lh];
    acc = wmma4(a, b, acc);
  }
  const float bv = bout[n0 + lh];
  for (int v = 0; v < 8; ++v) {
    const int mo = m0 + v + 8 * half;
    out_rep[(size_t)mo * VDIM + n0 + lh] = acc[v] + bv;
  }
}

// ---------------------------------------------------------------------------
extern "C" void kernel_launch(void* const* d_in, const int* in_sizes, int n_in,
                              void* d_out, int out_size, void* d_ws, size_t ws_size,
                              hipStream_t stream) {
  const float* value = (const float*)d_in[0];   // (8,256,64)
  const float* key_  = (const float*)d_in[1];   // (8,256,128)
  const float* query = (const float*)d_in[2];   // (1,64,128)
  const int*   mask  = (const int*)  d_in[3];   // (8,256,64)
  const float* Wq    = (const float*)d_in[4];   // (128,1024)
  const float* bq    = (const float*)d_in[5];   // (1024,)
  const float* Wk    = (const float*)d_in[6];   // (128,1024)
  const float* bk    = (const float*)d_in[7];   // (1024,)
  const float* Wout  = (const float*)d_in[8];   // (512,64)
  const float* bout  = (const float*)d_in[9];   // (64,)

  float* out_rep = (float*)d_out;                                  // 8*64*64
  float* p_out   = (float*)d_out + (size_t)NB * TQQ * VDIM;        // 8*8*64*256*64

  float* ws  = (float*)d_ws;
  float* hq  = ws;                                   // 64*1024      (256 KB)
  float* hk  = hq + (size_t)TQQ * NH * KQD;          // 8*256*1024   (8 MB)
  float* ows = hk + (size_t)NB * TKK * NH * KQD;     // 8*8*64*64    (1 MB)

  // hq: M=64, N=1024, K=128 -> 256 tiles -> 32 blocks of 8 waves
  proj_kernel<<<32, 256, 0, stream>>>(query, Wq, bq, hq, TQQ, NH * KQD, KQD);
  // hk: M=2048, N=1024, K=128 -> 8192 tiles -> 1024 blocks
  proj_kernel<<<1024, 256, 0, stream>>>(key_, Wk, bk, hk, NB * TKK, NH * KQD, KQD);
  // attention + p_attn stream: (q-chunk, head, batch)
  attn_kernel<<<dim3(4, NH, NB), 256, 0, stream>>>(hq, hk, mask, value, ows, p_out);
  // final projection: 128 tiles -> 16 blocks
  outproj_kernel<<<16, 256, 0, stream>>>(ows, Wout, bout, out_rep);
}